// TemporalGNN_27298812134106
// MI455X (gfx1250) — compile-verified
//
#include <hip/hip_runtime.h>
#include <hip/hip_bf16.h>
#include <math.h>

// ---------------- problem constants ----------------
#define NN      20000      // nodes
#define NE      320000     // edges
#define BB      4          // batch
#define FIN     16         // input features
#define HD      64         // hidden
#define PP      12         // periods
#define FP      (FIN*PP)   // 192 floats per (b,n)

typedef __attribute__((ext_vector_type(2))) float v2f;
typedef __attribute__((ext_vector_type(8))) float v8f;

// branchless fast activations (v_exp_f32 / v_rcp_f32, no EXEC divergence)
__device__ __forceinline__ float fast_sigmoid(float x) {
    return __builtin_amdgcn_rcpf(1.0f + __expf(-x));
}
__device__ __forceinline__ float fast_tanh(float x) {
    float t = __expf(-2.0f * fabsf(x));            // t in (0,1], never overflows
    float r = (1.0f - t) * __builtin_amdgcn_rcpf(1.0f + t);
    return copysignf(r, x);
}

// ---------------- stage 1: degree with self loops ----------------
__global__ void k_deg_init(float* __restrict__ deg) {
    int i = blockIdx.x * blockDim.x + threadIdx.x;
    if (i < NN) deg[i] = 1.0f;   // self-loop weight
}

__global__ void k_deg_edge(const long long* __restrict__ ei,
                           const float* __restrict__ ew,
                           float* __restrict__ deg) {
    int e = blockIdx.x * blockDim.x + threadIdx.x;
    if (e < NE) atomicAdd(&deg[(int)ei[NE + e]], ew[e]);
}

__global__ void k_dinv(const float* __restrict__ deg, float* __restrict__ dinv) {
    int i = blockIdx.x * blockDim.x + threadIdx.x;
    if (i < NN) {
        float d = deg[i];
        dinv[i] = (d > 0.0f) ? rsqrtf(d) : 0.0f;
    }
}

__global__ void k_norm(const long long* __restrict__ ei,
                       const float* __restrict__ ew,
                       const float* __restrict__ dinv,
                       float* __restrict__ nv) {
    int e = blockIdx.x * blockDim.x + threadIdx.x;
    if (e < NE) nv[e] = dinv[(int)ei[e]] * ew[e] * dinv[(int)ei[NE + e]];
}

// agg[b,n,:] starts at x[b,n,:] * dinv[n]^2 (self-loop contribution); float4 path
__global__ void k_agg_init(const float4* __restrict__ x4,
                           const float* __restrict__ dinv,
                           float4* __restrict__ agg4) {
    long long i = (long long)blockIdx.x * blockDim.x + threadIdx.x;
    if (i < (long long)BB * NN * (FP / 4)) {
        int n = (int)((i / (FP / 4)) % NN);
        float di = dinv[n];
        float s = di * di;
        float4 v = x4[i];
        v.x *= s; v.y *= s; v.z *= s; v.w *= s;
        agg4[i] = v;
    }
}

// one thread per (edge, float4 of features); loops over batch
__global__ void k_scatter(const float4* __restrict__ x4,
                          const long long* __restrict__ ei,
                          const float* __restrict__ nv,
                          float* __restrict__ agg) {
    long long tid = (long long)blockIdx.x * blockDim.x + threadIdx.x;
    if (tid >= (long long)NE * (FP / 4)) return;
    int e  = (int)(tid / (FP / 4));
    int f4 = (int)(tid % (FP / 4));
    int s  = (int)ei[e];
    int d  = (int)ei[NE + e];
    float nm = nv[e];
#pragma unroll
    for (int b = 0; b < BB; ++b) {
        float4 v = x4[((size_t)b * NN + s) * (FP / 4) + f4];
        float* dst = &agg[(((size_t)b * NN + d) * (FP / 4) + f4) * 4];
        atomicAdd(dst + 0, v.x * nm);
        atomicAdd(dst + 1, v.y * nm);
        atomicAdd(dst + 2, v.z * nm);
        atomicAdd(dst + 3, v.w * nm);
    }
}

// ---------------- stage 2: fold weights, biases, softmax ----------------
// Wz = Wg_z(16x64) @ Wl_z[:64](64x64); Wh likewise; bz = bl_z + bg_z@Wl_z[:64]; probs = softmax(att)
__global__ void k_prep(const float* __restrict__ Wg_z, const float* __restrict__ Wg_h,
                       const float* __restrict__ Wl_z, const float* __restrict__ bl_z,
                       const float* __restrict__ Wl_h, const float* __restrict__ bl_h,
                       const float* __restrict__ bg_z, const float* __restrict__ bg_h,
                       const float* __restrict__ att,
                       float* __restrict__ Wz, float* __restrict__ Wh,
                       float* __restrict__ bz, float* __restrict__ bh,
                       float* __restrict__ probs) {
    int t = blockIdx.x * blockDim.x + threadIdx.x;
    if (t < 1024) {
        int k = t >> 6, n = t & 63;
        float a = 0.0f;
        for (int j = 0; j < HD; ++j) a += Wg_z[k * HD + j] * Wl_z[j * HD + n];
        Wz[t] = a;
    } else if (t < 2048) {
        int u = t - 1024, k = u >> 6, n = u & 63;
        float a = 0.0f;
        for (int j = 0; j < HD; ++j) a += Wg_h[k * HD + j] * Wl_h[j * HD + n];
        Wh[u] = a;
    } else if (t < 2112) {
        int n = t - 2048;
        float a = bl_z[n];
        for (int j = 0; j < HD; ++j) a += bg_z[j] * Wl_z[j * HD + n];
        bz[n] = a;
    } else if (t < 2176) {
        int n = t - 2112;
        float a = bl_h[n];
        for (int j = 0; j < HD; ++j) a += bg_h[j] * Wl_h[j * HD + n];
        bh[n] = a;
    } else if (t < 2176 + PP) {
        int q = t - 2176;
        float m = att[0];
        for (int i = 1; i < PP; ++i) m = fmaxf(m, att[i]);
        float s = 0.0f;
        for (int i = 0; i < PP; ++i) s += expf(att[i] - m);
        probs[q] = expf(att[q] - m) / s;
    }
}

// ---------------- stage 3: per-node GRU (degenerate) via fp32 WMMA ----------------
// One wave owns 16 nodes of one batch. A = agg tile (16 nodes x 16 feat) per period,
// B = folded weights (16x64 in four 16x16 N-tiles). V_WMMA_F32_16X16X4_F32, K in chunks of 4.
__global__ __launch_bounds__(256) void k_gru_wmma(
        const float* __restrict__ agg,
        const float* __restrict__ Wz, const float* __restrict__ Wh,
        const float* __restrict__ bz, const float* __restrict__ bh,
        const float* __restrict__ probs,
        const float* __restrict__ Wout, const float* __restrict__ bout,
        float* __restrict__ out) {
    __shared__ float s_h[8][16][HD];   // 32 KB: per-wave 16x64 relu(Hacc) staging

    const int lane = threadIdx.x & 31;
    const int w    = threadIdx.x >> 5;
    const int wave = blockIdx.x * 8 + w;          // 625 blocks * 8 waves = 5000 tiles exactly
    const int b    = wave / (NN / 16);
    const int n0   = (wave % (NN / 16)) * 16;
    const int nlo  = lane & 15;
    const int hi   = lane >> 4;

    // B operands: lane(lo): K = 4kc+0,1 at N=nlo; lane(hi): K = 4kc+2,3. Row-major W[k*64+n].
    v2f Bz[4][4], Bh[4][4];
#pragma unroll
    for (int j = 0; j < 4; ++j) {
        const int col = 16 * j + nlo;
#pragma unroll
        for (int kc = 0; kc < 4; ++kc) {
            const int k0 = 4 * kc + 2 * hi;
            v2f tz; tz.x = Wz[k0 * HD + col]; tz.y = Wz[(k0 + 1) * HD + col];
            v2f th; th.x = Wh[k0 * HD + col]; th.y = Wh[(k0 + 1) * HD + col];
            Bz[j][kc] = tz; Bh[j][kc] = th;
        }
    }
    float pr[PP];
#pragma unroll
    for (int p = 0; p < PP; ++p) pr[p] = probs[p];
    float bzv[4], bhv[4];
#pragma unroll
    for (int j = 0; j < 4; ++j) { bzv[j] = bz[16 * j + nlo]; bhv[j] = bh[16 * j + nlo]; }

    float hacc[4][8];
#pragma unroll
    for (int j = 0; j < 4; ++j)
#pragma unroll
        for (int r = 0; r < 8; ++r) hacc[j][r] = 0.0f;

    // agg(b, n0+nlo, f, p): f-stride = PP floats
    const float* abase = agg + ((size_t)(b * NN + n0 + nlo) * FIN) * PP;

    for (int p = 0; p < PP; ++p) {
        v2f A[4];
#pragma unroll
        for (int kc = 0; kc < 4; ++kc) {
            const int k0 = 4 * kc + 2 * hi;
            v2f t; t.x = abase[k0 * PP + p]; t.y = abase[(k0 + 1) * PP + p];
            A[kc] = t;
        }
        const float pb = pr[p];
#pragma unroll
        for (int j = 0; j < 4; ++j) {
            v8f cz = {};
            v8f ch = {};
#pragma unroll
            for (int kc = 0; kc < 4; ++kc) {
                cz = __builtin_amdgcn_wmma_f32_16x16x4_f32(
                        false, A[kc], false, Bz[j][kc], (short)0, cz, false, false);
                ch = __builtin_amdgcn_wmma_f32_16x16x4_f32(
                        false, A[kc], false, Bh[j][kc], (short)0, ch, false, false);
            }
#pragma unroll
            for (int r = 0; r < 8; ++r) {
                float z  = fast_sigmoid(cz[r] + bzv[j]);
                float ht = fast_tanh(ch[r] + bhv[j]);
                hacc[j][r] += pb * (1.0f - z) * ht;   // Hn = (1-Z)*Ht since H0 = 0
            }
        }
    }

    // stage relu(Hacc) to LDS: D frag (j,r,lane) -> node m = 8*hi + r, col n = 16*j + nlo
#pragma unroll
    for (int j = 0; j < 4; ++j)
#pragma unroll
        for (int r = 0; r < 8; ++r)
            s_h[w][8 * hi + r][16 * j + nlo] = fmaxf(hacc[j][r], 0.0f);
    __syncthreads();

    // out[b, n0+m, q] = bout[q] + sum_h s_h[m][h]*Wout[h*12+q]; 192 outputs -> 6 per lane
#pragma unroll
    for (int t = 0; t < 6; ++t) {
        const int o = lane * 6 + t;
        const int m = o / PP, q = o % PP;
        float acc = bout[q];
#pragma unroll 8
        for (int h = 0; h < HD; ++h) acc += s_h[w][m][h] * Wout[h * PP + q];
        out[((size_t)(b * NN + n0 + m)) * PP + q] = acc;
    }
}

// ---------------- launcher ----------------
extern "C" void kernel_launch(void* const* d_in, const int* in_sizes, int n_in,
                              void* d_out, int out_size, void* d_ws, size_t ws_size,
                              hipStream_t stream) {
    const float*     x    = (const float*)d_in[0];
    const long long* ei   = (const long long*)d_in[1];
    const float*     ew   = (const float*)d_in[2];
    const float*     Wg_z = (const float*)d_in[3];
    const float*     bg_z = (const float*)d_in[4];
    // d_in[5] Wg_r, d_in[6] bg_r: dead (H0 = 0 kills the R gate)
    const float*     Wg_h = (const float*)d_in[7];
    const float*     bg_h = (const float*)d_in[8];
    const float*     Wl_z = (const float*)d_in[9];
    const float*     bl_z = (const float*)d_in[10];
    // d_in[11] Wl_r, d_in[12] bl_r: dead
    const float*     Wl_h = (const float*)d_in[13];
    const float*     bl_h = (const float*)d_in[14];
    const float*     att  = (const float*)d_in[15];
    const float*     Wout = (const float*)d_in[16];
    const float*     bout = (const float*)d_in[17];
    float*           out  = (float*)d_out;

    // workspace carve-up (floats)
    float* ws    = (float*)d_ws;
    float* agg   = ws;                          // BB*NN*FP = 15,360,000
    float* deg   = agg + (size_t)BB * NN * FP;  // 20,000
    float* dinv  = deg + NN;                    // 20,000
    float* normv = dinv + NN;                   // 320,000
    float* Wz    = normv + NE;                  // 1,024
    float* Wh    = Wz + FIN * HD;               // 1,024
    float* bz    = Wh + FIN * HD;               // 64
    float* bh    = bz + HD;                     // 64
    float* probs = bh + HD;                     // 12

    const int T = 256;
    k_deg_init<<<(NN + T - 1) / T, T, 0, stream>>>(deg);
    k_deg_edge<<<(NE + T - 1) / T, T, 0, stream>>>(ei, ew, deg);
    k_dinv<<<(NN + T - 1) / T, T, 0, stream>>>(deg, dinv);
    k_norm<<<(NE + T - 1) / T, T, 0, stream>>>(ei, ew, dinv, normv);

    long long aggTot4 = (long long)BB * NN * (FP / 4);
    k_agg_init<<<(unsigned)((aggTot4 + T - 1) / T), T, 0, stream>>>(
        (const float4*)x, dinv, (float4*)agg);

    long long scTot4 = (long long)NE * (FP / 4);
    k_scatter<<<(unsigned)((scTot4 + T - 1) / T), T, 0, stream>>>(
        (const float4*)x, ei, normv, agg);

    k_prep<<<(2176 + PP + T - 1) / T, T, 0, stream>>>(Wg_z, Wg_h, Wl_z, bl_z, Wl_h, bl_h,
                                                      bg_z, bg_h, att, Wz, Wh, bz, bh, probs);

    // 4 batches * 1250 node-tiles = 5000 waves; 8 waves per block
    k_gru_wmma<<<(BB * (NN / 16)) / 8, 256, 0, stream>>>(agg, Wz, Wh, bz, bh, probs,
                                                         Wout, bout, out);
}